// TotalLoss_33809982554776
// MI455X (gfx1250) — compile-verified
//
#include <hip/hip_runtime.h>

// ---------------------------------------------------------------------------
// Chamfer + Sinkhorn-EMD loss on 16 x [2048,3] fp32 point clouds, gfx1250.
//
// d2 tiles computed with V_WMMA_F32_16X16X4_F32 via cost embedding. Two
// orientations, chosen so the C operand is loop-invariant in each kernel:
//  chamfer (rows fixed per wave, cols sweep):
//     A=(-2x0,-2x1,-2x2,1)  B=(y0,y1,y2,|y|^2)  C[i][*]=|x_i|^2 (per row)
//  sinkhorn (cols fixed per wave, rows sweep):
//     A=(p0,p1,p2,|p|^2)    B=(-2q0,-2q1,-2q2,1) C[*][j]=|q_j|^2 (uniform)
// Either way D = |x|^2+|y|^2-2x.y = exact fp32 squared distance.
// The 256MB cost matrix is never materialized: points live in LDS pre-packed
// in the per-lane WMMA operand layout (float2[point][lane-half]); inner loops
// are branch-free: one ds_load_b64 per operand, WMMA, independent sqrt/exp2.
// ---------------------------------------------------------------------------

typedef __attribute__((ext_vector_type(2))) float v2f;
typedef __attribute__((ext_vector_type(8))) float v8f;

#define BATCH   16
#define NPTS    2048
#define NTILES  (NPTS / 16)
#define EPS_    0.05f
#define SCALE_  28.853900817779268f   // log2(e) / EPS
#define LN2_    0.6931471805599453f
#define LOGN_   7.6246189861593985f   // ln(2048)
#define NEG_BIG (-3.0e38f)
#define POS_BIG (3.0e38f)

__device__ __forceinline__ v8f wmma_d2(v2f a, v2f b, v8f c) {
  // D = A(16x4 f32) x B(4x16 f32) + C(16x16 f32)
  return __builtin_amdgcn_wmma_f32_16x16x4_f32(false, a, false, b,
                                               (short)0, c, false, false);
}

// ---------------------------------------------------------------------------
__global__ __launch_bounds__(256) void init_ws_kernel(float* __restrict__ f,
                                                      float* __restrict__ g,
                                                      unsigned* __restrict__ colmin,
                                                      float* __restrict__ rowpart) {
  int i = blockIdx.x * blockDim.x + threadIdx.x;
  if (i < BATCH * NPTS) {
    f[i] = 0.0f;
    g[i] = 0.0f;
    colmin[i] = 0x7F800000u;  // +inf bit pattern (stored values are >= 1e-12 > 0)
  }
  if (i < 256) rowpart[i] = 0.0f;
}

// ---------------------------------------------------------------------------
// Chamfer: grid (16 rowblocks, 16 batches), 256 threads = 8 waves.
// Wave w owns row tile (blockIdx.x*8 + w): 16 rows of X, sweeps all 128
// column tiles of Y. A and C operands are loop-invariant; per tile the body
// is: ds_load_b64 (B) -> WMMA -> min3 tree -> ds_min_u32. Row mins -> shfl
// butterfly -> per-block partial; col mins -> LDS then global uint atomicMin.
// ---------------------------------------------------------------------------
__global__ __launch_bounds__(256) void chamfer_kernel(const float* __restrict__ X,
                                                      const float* __restrict__ Y,
                                                      unsigned* __restrict__ colmin,
                                                      float* __restrict__ rowpart) {
  __shared__ float2   syB[NPTS * 2];  // per-half B operand: (y0,y1)|(y2,|y|^2)  32 KB
  __shared__ unsigned scol[NPTS];     // per-column running min (bits)            8 KB
  __shared__ float    swsum[16];

  const int b    = blockIdx.y;
  const int tid  = threadIdx.x;
  const int lane = tid & 31;
  const int wave = tid >> 5;
  const int half = lane >> 4;        // 0: rows v+0..7 ; 1: rows v+8..15
  const size_t pbase = (size_t)b * NPTS;

  for (int idx = tid; idx < NPTS; idx += 256) {
    const float* yp = Y + (pbase + idx) * 3;
    float y0 = yp[0], y1 = yp[1], y2 = yp[2];
    syB[idx * 2 + 0] = make_float2(y0, y1);
    syB[idx * 2 + 1] = make_float2(y2, y0 * y0 + y1 * y1 + y2 * y2);
    scol[idx] = 0x7F800000u;
  }
  __syncthreads();

  // A matrix: fixed 16 rows of X for this wave (built once, branchless).
  // 32-bit A 16x4 layout: lanes 0-15 hold K={0,1}, lanes 16-31 hold K={2,3}.
  const int rbase = (blockIdx.x * 8 + wave) * 16;
  const float* xp = X + (pbase + rbase + (lane & 15)) * 3;
  float x0 = xp[0], x1 = xp[1], x2 = xp[2];
  float xsum = x0 * x0 + x1 * x1 + x2 * x2;   // lane r holds |x_{rbase+r}|^2
  v2f amat;
  amat[0] = half ? (-2.0f * x2) : (-2.0f * x0);
  amat[1] = half ? 1.0f         : (-2.0f * x1);

  // C matrix: |x_i|^2 per row. Lane in half h needs rows rbase+8h+v (v=0..7):
  // gather from the lane holding that row via shfl (once per wave).
  v8f cvec;
#pragma unroll
  for (int v = 0; v < 8; ++v) cvec[v] = __shfl(xsum, 8 * half + v, 16);

  float rmin[8];
#pragma unroll
  for (int v = 0; v < 8; ++v) rmin[v] = POS_BIG;

  for (int ct = 0; ct < NTILES; ++ct) {
    const int j = ct * 16 + (lane & 15);
    float2 bq = syB[j * 2 + half];   // one ds_load_b64, lane-dependent, no branch
    v2f bmat; bmat[0] = bq.x; bmat[1] = bq.y;

    v8f d2 = wmma_d2(amat, bmat, cvec);

    float cm = fminf(fminf(fminf(d2[0], d2[1]), fminf(d2[2], d2[3])),
                     fminf(fminf(d2[4], d2[5]), fminf(d2[6], d2[7])));
#pragma unroll
    for (int v = 0; v < 8; ++v) rmin[v] = fminf(rmin[v], d2[v]);
    // Clamp before bit-min (reference clamps d2 at 1e-12; also keeps bits >= 0
    // so uint ordering is valid). Lanes l and l+16 hit the same word: the DS
    // atomic unit merges the two 8-row mins into the 16-row column min.
    cm = fmaxf(cm, 1e-12f);
    atomicMin(&scol[j], __float_as_uint(cm));
  }

  // Row mins: for fixed VGPR v the 16 columns of a tile live across the 16
  // lanes of the half -> butterfly min within each group of 16 lanes.
#pragma unroll
  for (int off = 1; off < 16; off <<= 1) {
#pragma unroll
    for (int v = 0; v < 8; ++v)
      rmin[v] = fminf(rmin[v], __shfl_xor(rmin[v], off, 16));
  }
  if ((lane & 15) == 0) {
    float s = 0.0f;
#pragma unroll
    for (int v = 0; v < 8; ++v) s += fmaxf(rmin[v], 1e-12f);  // clamp-after-min == min-of-clamped
    swsum[wave * 2 + half] = s;   // half 0: rows +0..7, half 1: rows +8..15
  }
  __syncthreads();
  if (tid == 0) {
    float s = 0.0f;
    for (int k = 0; k < 16; ++k) s += swsum[k];
    rowpart[blockIdx.y * 16 + blockIdx.x] = s;
  }
  for (int j = tid; j < NPTS; j += 256)
    atomicMin(&colmin[pbase + j], scol[j]);
}

// ---------------------------------------------------------------------------
// One Sinkhorn half-step:  out[j] = -eps*( LSE_i[(in[i]-C(i,j))/eps] - ln N )
// where C(i,j) = ||P_i - Q_j||.  g-update: P=X,in=f,Q=Y ; f-update: P=Y,in=g,Q=X.
// Grid (16 colblocks, 16 batches), 256 threads = 8 waves; each wave owns a
// 16-column tile and streams 128 row tiles. Per tile: branch-free operand
// loads, WMMA, 8 independent sqrt+exp2 (tree-max shifted), one (m,s) merge.
// ---------------------------------------------------------------------------
__global__ __launch_bounds__(256) void sinkhorn_kernel(const float* __restrict__ P,
                                                       const float* __restrict__ Q,
                                                       const float* __restrict__ in_pot,
                                                       float* __restrict__ out_pot) {
  __shared__ float2 sPA[NPTS * 2];  // per-half A operand: (p0,p1)|(p2,|p|^2)  32 KB
  __shared__ float  sF[NPTS];       // incoming potential                      8 KB

  const int b    = blockIdx.y;
  const int tid  = threadIdx.x;
  const int lane = tid & 31;
  const int wave = tid >> 5;
  const int half = lane >> 4;
  const size_t pbase = (size_t)b * NPTS;

  for (int idx = tid; idx < NPTS; idx += 256) {
    const float* pp = P + (pbase + idx) * 3;
    float p0 = pp[0], p1 = pp[1], p2 = pp[2];
    sPA[idx * 2 + 0] = make_float2(p0, p1);
    sPA[idx * 2 + 1] = make_float2(p2, p0 * p0 + p1 * p1 + p2 * p2);
    sF[idx] = in_pot[pbase + idx];
  }
  __syncthreads();

  // B matrix + C broadcast: fixed 16 columns (Q points), built once, branchless.
  const int j = (blockIdx.x * 8 + wave) * 16 + (lane & 15);
  const float* qp = Q + (pbase + j) * 3;
  float q0 = qp[0], q1 = qp[1], q2 = qp[2];
  v2f bmat;
  bmat[0] = half ? (-2.0f * q2) : (-2.0f * q0);
  bmat[1] = half ? 1.0f         : (-2.0f * q1);
  const float qsum = q0 * q0 + q1 * q1 + q2 * q2;
  v8f cvec;
#pragma unroll
  for (int v = 0; v < 8; ++v) cvec[v] = qsum;

  float m = NEG_BIG, s = 0.0f;   // streaming LSE state (base-2 domain)

  for (int rt = 0; rt < NTILES; ++rt) {
    const int rbase = rt * 16;
    float2 pv = sPA[(rbase + (lane & 15)) * 2 + half];  // one ds_load_b64
    v2f amat; amat[0] = pv.x; amat[1] = pv.y;

    // Rows covered by this lane: rbase + v + 8*half (v=0..7) -> 8 contiguous f's.
    const float4 fA = *(const float4*)&sF[rbase + 8 * half];
    const float4 fB = *(const float4*)&sF[rbase + 8 * half + 4];
    const float fv[8] = {fA.x, fA.y, fA.z, fA.w, fB.x, fB.y, fB.z, fB.w};

    v8f d2 = wmma_d2(amat, bmat, cvec);

    float u[8];
#pragma unroll
    for (int v = 0; v < 8; ++v) {
      float d = fmaxf(d2[v], 1e-12f);
      float c = sqrtf(d);                 // C_ij (p=1 cost)
      u[v] = (fv[v] - c) * SCALE_;        // (f_i - C_ij)/eps in log2 units
    }
    // Tree max (v_max3), then 8 independent exp2, then one streaming merge.
    float tm = fmaxf(fmaxf(fmaxf(u[0], u[1]), fmaxf(u[2], u[3])),
                     fmaxf(fmaxf(u[4], u[5]), fmaxf(u[6], u[7])));
    float st = 0.0f;
#pragma unroll
    for (int v = 0; v < 8; ++v) st += exp2f(u[v] - tm);

    float nm = fmaxf(m, tm);
    s = s * exp2f(m - nm) + st * exp2f(tm - nm);
    m = nm;
  }

  // Merge lane pair (rows 0-7 half vs rows 8-15 half of every tile).
  float om = __shfl_xor(m, 16, 32);
  float os = __shfl_xor(s, 16, 32);
  float nm = fmaxf(m, om);
  s = s * exp2f(m - nm) + os * exp2f(om - nm);
  m = nm;

  float res = -EPS_ * ((m + log2f(s)) * LN2_ - LOGN_);
  if (half == 0) out_pot[pbase + j] = res;
}

// ---------------------------------------------------------------------------
// total = sum(rowmins) + sum(colmins) + sum(f) + sum(g)
//   (emd_loss = sum_b N*(mean f + mean g) = sum f + sum g since N == M)
// ---------------------------------------------------------------------------
__global__ __launch_bounds__(256) void finalize_kernel(const float* __restrict__ fg,
                                                       const unsigned* __restrict__ colmin,
                                                       const float* __restrict__ rowpart,
                                                       float* __restrict__ out) {
  __shared__ float red[256];
  const int tid = threadIdx.x;
  float acc = 0.0f;
  for (int i = tid; i < 2 * BATCH * NPTS; i += 256) acc += fg[i];             // f then g
  for (int i = tid; i < BATCH * NPTS; i += 256) acc += __uint_as_float(colmin[i]);
  for (int i = tid; i < 256; i += 256) acc += rowpart[i];
  red[tid] = acc;
  __syncthreads();
  for (int s = 128; s > 0; s >>= 1) {
    if (tid < s) red[tid] += red[tid + s];
    __syncthreads();
  }
  if (tid == 0) out[0] = red[0];
}

// ---------------------------------------------------------------------------
extern "C" void kernel_launch(void* const* d_in, const int* in_sizes, int n_in,
                              void* d_out, int out_size, void* d_ws, size_t ws_size,
                              hipStream_t stream) {
  (void)in_sizes; (void)n_in; (void)out_size; (void)ws_size;
  const float* X = (const float*)d_in[0];   // output  [16,2048,3] f32
  const float* Y = (const float*)d_in[1];   // target  [16,2048,3] f32
  float* out = (float*)d_out;

  // Workspace layout (floats): f[32768] | g[32768] | colmin[32768] | rowpart[256]
  float*    f       = (float*)d_ws;
  float*    g       = f + BATCH * NPTS;
  unsigned* colmin  = (unsigned*)(g + BATCH * NPTS);
  float*    rowpart = (float*)(colmin + BATCH * NPTS);

  init_ws_kernel<<<dim3((BATCH * NPTS + 255) / 256), dim3(256), 0, stream>>>(
      f, g, colmin, rowpart);

  chamfer_kernel<<<dim3(16, BATCH), dim3(256), 0, stream>>>(X, Y, colmin, rowpart);

  for (int it = 0; it < 10; ++it) {
    // g_j update: reduce over rows i of X using old f
    sinkhorn_kernel<<<dim3(16, BATCH), dim3(256), 0, stream>>>(X, Y, f, g);
    // f_i update: reduce over rows j of Y using new g (d2 construction is symmetric)
    sinkhorn_kernel<<<dim3(16, BATCH), dim3(256), 0, stream>>>(Y, X, g, f);
  }

  finalize_kernel<<<dim3(1), dim3(256), 0, stream>>>(f, colmin, rowpart, out);
}